// VMGLinear_27504970564165
// MI455X (gfx1250) — compile-verified
//
#include <hip/hip_runtime.h>

typedef __attribute__((ext_vector_type(2))) float v2f;
typedef __attribute__((ext_vector_type(4))) float v4f;
typedef __attribute__((ext_vector_type(8))) float v8f;

#define M_SAMPLES 1024
#define IN_DIM    512
#define RDIM      513   // IN_DIM + bias row
#define CDIM      512

// ---------------------------------------------------------------------------
// Kernel 1: h1 = xa @ mu  via V_WMMA_F32_16X16X4_F32  (f32 matrix pipe)
// One wave (32 threads) computes one 16x16 output tile, K = 513 padded to 516.
// A layout (16x4, 2 VGPRs): lanes 0-15 hold K=k+0/k+1, lanes 16-31 K=k+2/k+3.
// B layout (4x16, 2 VGPRs): VGPR0 = row k+koff, VGPR1 = row k+koff+1, N=lane%16.
// D layout (16x16, 8 VGPRs): VGPR v -> (M = m0 + v + 8*hi, N = c0 + lane%16).
// ---------------------------------------------------------------------------
__global__ __launch_bounds__(32) void vmg_gemm_wmma(const float* __restrict__ x,
                                                    const float* __restrict__ mu,
                                                    float* __restrict__ out) {
    const int lane = threadIdx.x;     // 0..31
    const int ln   = lane & 15;
    const int hi   = lane >> 4;       // 0 or 1
    const int m0   = blockIdx.y << 4;
    const int c0   = blockIdx.x << 4;

    const int am   = m0 + ln;         // A-matrix row handled by this lane
    const int bn   = c0 + ln;         // B/D column handled by this lane
    const int koff = hi * 2;

    const float* __restrict__ xrow = x + (size_t)am * IN_DIM;

    v8f acc = {};

    // Main K loop: all indices < 512, no bias/padding guards needed.
    for (int k = 0; k < IN_DIM; k += 4) {
        v2f a, b;
        a.x = xrow[k + koff];
        a.y = xrow[k + koff + 1];
        b.x = mu[(size_t)(k + koff) * CDIM + bn];
        b.y = mu[(size_t)(k + koff + 1) * CDIM + bn];
        acc = __builtin_amdgcn_wmma_f32_16x16x4_f32(
            /*neg_a=*/false, a, /*neg_b=*/false, b,
            /*c_mod=*/(short)0, acc, /*reuse_a=*/false, /*reuse_b=*/false);
    }
    // Tail step k = 512..515: xa[:,512] = 1 (bias), rows >= 513 are zero pad.
    {
        const int k0 = IN_DIM + koff;
        const int k1 = IN_DIM + koff + 1;
        v2f a, b;
        a.x = (k0 == IN_DIM) ? 1.0f : 0.0f;
        a.y = (k1 == IN_DIM) ? 1.0f : 0.0f;
        b.x = (k0 < RDIM) ? mu[(size_t)k0 * CDIM + bn] : 0.0f;
        b.y = (k1 < RDIM) ? mu[(size_t)k1 * CDIM + bn] : 0.0f;
        acc = __builtin_amdgcn_wmma_f32_16x16x4_f32(
            false, a, false, b, (short)0, acc, false, false);
    }

#pragma unroll
    for (int v = 0; v < 8; ++v) {
        out[(size_t)(m0 + v + hi * 8) * CDIM + bn] = acc[v];
    }
}

// ---------------------------------------------------------------------------
// Kernel 2: stream E once (1.07 GB -> the whole roofline), add sigma-scaled
// per-sample reduction onto the GEMM partial already sitting in `out`.
// One block (128 threads / 4 waves) per sample; thread t owns cols 4t..4t+3.
// a[r] = xa[m,r]*exp(0.5*logvar_in[r]) cached in LDS (broadcast, no bank
// conflicts). E is read with b128 non-temporal loads (512 B per wave-instr,
// single-use data bypasses the 192 MB L2); unroll-8 keeps ~8 b128 loads in
// flight per wave to cover HBM latency at 23.3 TB/s.
// ---------------------------------------------------------------------------
__global__ __launch_bounds__(128) void vmg_estream(const float* __restrict__ x,
                                                   const float* __restrict__ lv_in,
                                                   const float* __restrict__ lv_out,
                                                   const float* __restrict__ E,
                                                   float* __restrict__ out) {
    __shared__ float a[RDIM];
    const int m = blockIdx.x;
    const int t = threadIdx.x;                 // 0..127, covers 512 cols as v4f

    for (int r = t; r < RDIM; r += 128) {
        float xv = (r < IN_DIM) ? x[(size_t)m * IN_DIM + r] : 1.0f;
        a[r] = xv * __expf(0.5f * lv_in[r]);
    }
    __syncthreads();

    const v4f* __restrict__ Em =
        reinterpret_cast<const v4f*>(E + (size_t)m * RDIM * CDIM) + t;

    v4f acc = {0.0f, 0.0f, 0.0f, 0.0f};
#pragma unroll 8
    for (int r = 0; r < RDIM; ++r) {
        v4f e = __builtin_nontemporal_load(Em + (size_t)r * (CDIM / 4));
        float ar = a[r];
        acc.x = fmaf(ar, e.x, acc.x);
        acc.y = fmaf(ar, e.y, acc.y);
        acc.z = fmaf(ar, e.z, acc.z);
        acc.w = fmaf(ar, e.w, acc.w);
    }

    const int c = t * 4;
    float* o = out + (size_t)m * CDIM + c;
    o[0] = o[0] + __expf(0.5f * lv_out[c])     * acc.x;  // GEMM partial from k1
    o[1] = o[1] + __expf(0.5f * lv_out[c + 1]) * acc.y;
    o[2] = o[2] + __expf(0.5f * lv_out[c + 2]) * acc.z;
    o[3] = o[3] + __expf(0.5f * lv_out[c + 3]) * acc.w;
}

// ---------------------------------------------------------------------------
// Kernel 3: D_KL scalar (five small reductions, one block).
// ---------------------------------------------------------------------------
__global__ __launch_bounds__(256) void vmg_dkl(const float* __restrict__ mu,
                                               const float* __restrict__ lv_in,
                                               const float* __restrict__ lv_out,
                                               float* __restrict__ out) {
    const int t = threadIdx.x;
    float s_var_r = 0.0f, s_lv_in = 0.0f, s_var_c = 0.0f, s_lv_out = 0.0f, s_mu2 = 0.0f;

    for (int r = t; r < RDIM; r += 256) {
        float v = lv_in[r];
        s_var_r += __expf(v);
        s_lv_in += v;
    }
    for (int c = t; c < CDIM; c += 256) {
        float v = lv_out[c];
        s_var_c  += __expf(v);
        s_lv_out += v;
    }
    for (int i = t; i < RDIM * CDIM; i += 256) {
        float v = mu[i];
        s_mu2 = fmaf(v, v, s_mu2);
    }

    __shared__ float red[5][256];
    red[0][t] = s_var_r;  red[1][t] = s_lv_in;  red[2][t] = s_var_c;
    red[3][t] = s_lv_out; red[4][t] = s_mu2;
    __syncthreads();
    for (int s = 128; s > 0; s >>= 1) {
        if (t < s) {
#pragma unroll
            for (int j = 0; j < 5; ++j) red[j][t] += red[j][t + s];
        }
        __syncthreads();
    }
    if (t == 0) {
        float dkl = 0.5f * (red[0][0] * red[2][0] + red[4][0]
                            - (float)(RDIM * CDIM)
                            - (float)CDIM * red[1][0]
                            - (float)RDIM * red[3][0]);
        out[(size_t)M_SAMPLES * CDIM] = dkl;
    }
}

// ---------------------------------------------------------------------------
extern "C" void kernel_launch(void* const* d_in, const int* in_sizes, int n_in,
                              void* d_out, int out_size, void* d_ws, size_t ws_size,
                              hipStream_t stream) {
    (void)in_sizes; (void)n_in; (void)out_size; (void)d_ws; (void)ws_size;

    const float* x      = (const float*)d_in[0];  // (1024, 512)
    const float* mu     = (const float*)d_in[1];  // (513, 512)
    const float* lv_in  = (const float*)d_in[2];  // (513,)
    const float* lv_out = (const float*)d_in[3];  // (512,)
    const float* E      = (const float*)d_in[4];  // (1024, 513, 512)
    float*       out    = (float*)d_out;          // 1024*512 h + 1 D_KL

    // 1) xa @ mu via f32 WMMA -> out[0 .. M*C)
    dim3 gridB(CDIM / 16, M_SAMPLES / 16);        // 32 x 64 tiles, 1 wave each
    vmg_gemm_wmma<<<gridB, 32, 0, stream>>>(x, mu, out);

    // 2) stream E (b128 NT), accumulate sigma-scaled term onto the GEMM partial
    vmg_estream<<<M_SAMPLES, 128, 0, stream>>>(x, lv_in, lv_out, E, out);

    // 3) D_KL scalar
    vmg_dkl<<<1, 256, 0, stream>>>(mu, lv_in, lv_out, out);
}